// MinimizeEnergy_85031762526429
// MI455X (gfx1250) — compile-verified
//
#include <hip/hip_runtime.h>
#include <hip/hip_bf16.h>

typedef float v2f __attribute__((ext_vector_type(2)));
typedef float v8f __attribute__((ext_vector_type(8)));

#define BLOCK 256
#define ESLOTS 256   // spread slots per energy term to avoid same-address atomics

// ---------------------------------------------------------------------------
// Wave32 partial sum via V_WMMA_F32_16X16X4_F32 (CDNA5 XDL path).
// A(16x4): lane L<16 holds A[L][0]=e; lane L>=16 holds A[L-16][2]=e.
// B = ones -> D[m][n] = e_m + e_{m+16}.  Each lane sums its 8 D rows:
// lane 0 ends with sum(e0..7,e16..23), lane 16 with sum(e8..15,e24..31).
// Lanes 0 and 16 together hold the full wave sum -> both push to LDS.
// EXEC must be all ones here: callers never early-return before this.
// ---------------------------------------------------------------------------
__device__ __forceinline__ float wave_partial_wmma(float e) {
  v2f a; a[0] = e;    a[1] = 0.0f;
  v2f b; b[0] = 1.0f; b[1] = 1.0f;
  v8f c = {};
  v8f d = __builtin_amdgcn_wmma_f32_16x16x4_f32(
      /*neg_a=*/false, a, /*neg_b=*/false, b,
      /*c_mod=*/(short)0, c, /*reuse_a=*/false, /*reuse_b=*/false);
  return d[0] + d[1] + d[2] + d[3] + d[4] + d[5] + d[6] + d[7];
}

// Block-level energy fold: WMMA wave partials -> LDS atomic -> one global
// atomic per block into a spread slot (kills same-address serialization).
__device__ __forceinline__ void block_energy_fold(float e, float* slot_base,
                                                  float* s_e) {
  if (threadIdx.x == 0) *s_e = 0.0f;
  __syncthreads();
  float s = wave_partial_wmma(e);          // EXEC all ones here
  if ((threadIdx.x & 15) == 0)             // lanes 0 and 16 of each wave
    atomicAdd(s_e, s);
  __syncthreads();
  if (threadIdx.x == 0)
    atomicAdd(&slot_base[blockIdx.x & (ESLOTS - 1)], *s_e);
}

__device__ __forceinline__ void load3(const float* __restrict__ p, int i,
                                      float& x, float& y, float& z) {
  const float* q = p + 3 * i;
  x = q[0]; y = q[1]; z = q[2];
}

// ---------------------------------------------------------------------------
// Bonds: E += relu((|p0-p1| - eq)^2 - tol^2)  (energy scale at finalize;
// gradient scale Kg = 1000/Nb folded here)
// ---------------------------------------------------------------------------
__global__ void __launch_bounds__(BLOCK)
k_bonds(const float* __restrict__ pos, const int* __restrict__ idcs,
        const float* __restrict__ eqv, const float* __restrict__ tol,
        float* __restrict__ gacc, float* __restrict__ esum, int n, float Kg) {
  __shared__ float s_e;
  int gid = blockIdx.x * BLOCK + threadIdx.x;
  float e = 0.0f;
  if (gid < n) {
    int i0 = idcs[2 * gid + 0], i1 = idcs[2 * gid + 1];
    float x0, y0, z0, x1, y1, z1;
    load3(pos, i0, x0, y0, z0);
    load3(pos, i1, x1, y1, z1);
    float dx = x0 - x1, dy = y0 - y1, dz = z0 - z1;
    float r = sqrtf(dx * dx + dy * dy + dz * dz);
    float d = r - eqv[gid];
    float t = tol[gid];
    float act = d * d - t * t;
    if (act > 0.0f) {
      e = act;
      float cg = Kg * 2.0f * d / r;
      float gx = cg * dx, gy = cg * dy, gz = cg * dz;
      atomicAdd(&gacc[3 * i0 + 0], gx);
      atomicAdd(&gacc[3 * i0 + 1], gy);
      atomicAdd(&gacc[3 * i0 + 2], gz);
      atomicAdd(&gacc[3 * i1 + 0], -gx);
      atomicAdd(&gacc[3 * i1 + 1], -gy);
      atomicAdd(&gacc[3 * i1 + 2], -gz);
    }
  }
  block_energy_fold(e, esum, &s_e);
}

// ---------------------------------------------------------------------------
// Angles: theta = acos(clip(cos)), E += relu((theta-eq)^2 - tol^2)
// d theta/d cos = -1/sqrt(1-cos^2); clip kills gradient for |cos|>=1.
// ---------------------------------------------------------------------------
__global__ void __launch_bounds__(BLOCK)
k_angles(const float* __restrict__ pos, const int* __restrict__ idcs,
         const float* __restrict__ eqv, const float* __restrict__ tol,
         float* __restrict__ gacc, float* __restrict__ esum, int n, float Kg) {
  __shared__ float s_e;
  int gid = blockIdx.x * BLOCK + threadIdx.x;
  float e = 0.0f;
  if (gid < n) {
    int i0 = idcs[3 * gid + 0], i1 = idcs[3 * gid + 1], i2 = idcs[3 * gid + 2];
    float x0, y0, z0, x1, y1, z1, x2, y2, z2;
    load3(pos, i0, x0, y0, z0);
    load3(pos, i1, x1, y1, z1);
    load3(pos, i2, x2, y2, z2);
    float ax = x0 - x1, ay = y0 - y1, az = z0 - z1;   // b0
    float bx = x2 - x1, by = y2 - y1, bz = z2 - z1;   // b1
    float a2 = ax * ax + ay * ay + az * az;
    float b2 = bx * bx + by * by + bz * bz;
    float dotv = ax * bx + ay * by + az * bz;
    float inv = rsqrtf(a2 * b2);
    float cosv = dotv * inv;
    float cosc = fminf(1.0f, fmaxf(-1.0f, cosv));
    float th = acosf(cosc);
    float d = th - eqv[gid];
    float t = tol[gid];
    float act = d * d - t * t;
    if (act > 0.0f) {
      e = act;
      if (fabsf(cosv) < 1.0f) {
        float dthdcos = -rsqrtf(1.0f - cosc * cosc);
        float Kc = Kg * 2.0f * d * dthdcos;           // dE/dcos
        float ca = cosv / a2, cb = cosv / b2;
        float g0x = Kc * (bx * inv - ax * ca);
        float g0y = Kc * (by * inv - ay * ca);
        float g0z = Kc * (bz * inv - az * ca);
        float g2x = Kc * (ax * inv - bx * cb);
        float g2y = Kc * (ay * inv - by * cb);
        float g2z = Kc * (az * inv - bz * cb);
        atomicAdd(&gacc[3 * i0 + 0], g0x);
        atomicAdd(&gacc[3 * i0 + 1], g0y);
        atomicAdd(&gacc[3 * i0 + 2], g0z);
        atomicAdd(&gacc[3 * i2 + 0], g2x);
        atomicAdd(&gacc[3 * i2 + 1], g2y);
        atomicAdd(&gacc[3 * i2 + 2], g2z);
        atomicAdd(&gacc[3 * i1 + 0], -(g0x + g2x));
        atomicAdd(&gacc[3 * i1 + 1], -(g0y + g2y));
        atomicAdd(&gacc[3 * i1 + 2], -(g0z + g2z));
      }
    }
  }
  block_energy_fold(e, esum + ESLOTS, &s_e);
}

// ---------------------------------------------------------------------------
// Dihedrals.  Reference phi = atan2(y,x) (praxeolitic) == atan2(-|s|*(u.n2),
// -(n1.n2)) with u=p0-p1, s=p2-p1, t=p3-p2, n1=u x s, n2=s x t.
// Energy term 2+cos(D-pi)+sin(D-pi/2) == 2-2cos(D);  dE/dphi = 2*Kg*sin(D).
// Blondel-Karplus gradients: gF=(|s|/|n1|^2)n1, gH=(|s|/|n2|^2)n2,
// dphi/dp0=gF, dphi/dp3=gH, dphi/dp1=(p-1)gF+q*gH, dphi/dp2=-p*gF-(q+1)gH,
// p=(u.s)/|s|^2, q=(t.s)/|s|^2.  (Signs verified against reference convention.)
// ---------------------------------------------------------------------------
__global__ void __launch_bounds__(BLOCK)
k_dihs(const float* __restrict__ pos, const int* __restrict__ idcs,
       const float* __restrict__ eqv,
       float* __restrict__ gacc, float* __restrict__ esum, int n, float Kg) {
  __shared__ float s_e;
  int gid = blockIdx.x * BLOCK + threadIdx.x;
  float e = 0.0f;
  if (gid < n) {
    int i0 = idcs[4 * gid + 0], i1 = idcs[4 * gid + 1];
    int i2 = idcs[4 * gid + 2], i3 = idcs[4 * gid + 3];
    float x0, y0, z0, x1, y1, z1, x2, y2, z2, x3, y3, z3;
    load3(pos, i0, x0, y0, z0);
    load3(pos, i1, x1, y1, z1);
    load3(pos, i2, x2, y2, z2);
    load3(pos, i3, x3, y3, z3);
    float ux = x0 - x1, uy = y0 - y1, uz = z0 - z1;   // b0
    float sx = x2 - x1, sy = y2 - y1, sz = z2 - z1;   // b1 (axis)
    float tx = x3 - x2, ty = y3 - y2, tz = z3 - z2;   // b2
    float n1x = uy * sz - uz * sy, n1y = uz * sx - ux * sz, n1z = ux * sy - uy * sx;
    float n2x = sy * tz - sz * ty, n2y = sz * tx - sx * tz, n2z = sx * ty - sy * tx;
    float s2 = sx * sx + sy * sy + sz * sz;
    float slen = sqrtf(s2);
    float X = -(n1x * n2x + n1y * n2y + n1z * n2z);
    float Y = -slen * (ux * n2x + uy * n2y + uz * n2z);
    float phi = atan2f(Y, X);
    float del = phi - eqv[gid];
    float cd, sd;
    __sincosf(del, &sd, &cd);
    e = 2.0f - 2.0f * cd;
    float dE = Kg * 2.0f * sd;                         // dE/dphi (scaled)
    float cF = dE * slen / (n1x * n1x + n1y * n1y + n1z * n1z);
    float cH = dE * slen / (n2x * n2x + n2y * n2y + n2z * n2z);
    float gFx = cF * n1x, gFy = cF * n1y, gFz = cF * n1z;
    float gHx = cH * n2x, gHy = cH * n2y, gHz = cH * n2z;
    float pp = (ux * sx + uy * sy + uz * sz) / s2;
    float qq = (tx * sx + ty * sy + tz * sz) / s2;
    float g1x = (pp - 1.0f) * gFx + qq * gHx;
    float g1y = (pp - 1.0f) * gFy + qq * gHy;
    float g1z = (pp - 1.0f) * gFz + qq * gHz;
    float g2x = -pp * gFx - (qq + 1.0f) * gHx;
    float g2y = -pp * gFy - (qq + 1.0f) * gHy;
    float g2z = -pp * gFz - (qq + 1.0f) * gHz;
    atomicAdd(&gacc[3 * i0 + 0], gFx);
    atomicAdd(&gacc[3 * i0 + 1], gFy);
    atomicAdd(&gacc[3 * i0 + 2], gFz);
    atomicAdd(&gacc[3 * i1 + 0], g1x);
    atomicAdd(&gacc[3 * i1 + 1], g1y);
    atomicAdd(&gacc[3 * i1 + 2], g1z);
    atomicAdd(&gacc[3 * i2 + 0], g2x);
    atomicAdd(&gacc[3 * i2 + 1], g2y);
    atomicAdd(&gacc[3 * i2 + 2], g2z);
    atomicAdd(&gacc[3 * i3 + 0], gHx);
    atomicAdd(&gacc[3 * i3 + 1], gHy);
    atomicAdd(&gacc[3 * i3 + 2], gHz);
  }
  block_energy_fold(e, esum + 2 * ESLOTS, &s_e);
}

// ---------------------------------------------------------------------------
// Finalize: gradient = nan_to_num(-gacc), clip per-atom norm to 0.1/dtau,
// write gradient (in place over accumulator region) + new_pos = pos + energy.
// ---------------------------------------------------------------------------
__device__ __forceinline__ float fixup(float v) {
  if (__builtin_isnan(v)) return 0.0f;
  if (__builtin_isinf(v)) return v > 0.0f ? 3.4028235e38f : -3.4028235e38f;
  return v;
}

__global__ void __launch_bounds__(BLOCK)
k_finalize(const float* __restrict__ pos, const float* __restrict__ esum,
           const float* __restrict__ dtau, float* __restrict__ out,
           int natoms, float bK, float aK, float dK) {
  int i = blockIdx.x * BLOCK + threadIdx.x;
  if (i < natoms) {
    float* gout = out + 3 * natoms;
    float gx = fixup(-gout[3 * i + 0]);
    float gy = fixup(-gout[3 * i + 1]);
    float gz = fixup(-gout[3 * i + 2]);
    float fn = sqrtf(gx * gx + gy * gy + gz * gz);
    float thr = 0.1f / dtau[0];
    float sc = (fn > thr) ? thr / fmaxf(fn, 1e-20f) : 1.0f;
    gout[3 * i + 0] = gx * sc;
    gout[3 * i + 1] = gy * sc;
    gout[3 * i + 2] = gz * sc;
    out[3 * i + 0] = pos[3 * i + 0];
    out[3 * i + 1] = pos[3 * i + 1];
    out[3 * i + 2] = pos[3 * i + 2];
    if (i == 0) {
      float sb = 0.0f, sa = 0.0f, sd = 0.0f;
      for (int j = 0; j < ESLOTS; ++j) {
        sb += esum[j];
        sa += esum[ESLOTS + j];
        sd += esum[2 * ESLOTS + j];
      }
      out[6 * natoms] = bK * sb + aK * sa + dK * sd;
    }
  }
}

// Movable update (duplicate indices write identical values -> benign).
__global__ void __launch_bounds__(BLOCK)
k_move(const float* __restrict__ pos, const int* __restrict__ mov,
       const float* __restrict__ dtau, float* __restrict__ out,
       int nmov, int natoms) {
  int j = blockIdx.x * BLOCK + threadIdx.x;
  if (j < nmov) {
    int idx = mov[j];
    float dt = dtau[0];
    const float* g = out + 3 * natoms + 3 * idx;
    out[3 * idx + 0] = pos[3 * idx + 0] + g[0] * dt;
    out[3 * idx + 1] = pos[3 * idx + 1] + g[1] * dt;
    out[3 * idx + 2] = pos[3 * idx + 2] + g[2] * dt;
  }
}

extern "C" void kernel_launch(void* const* d_in, const int* in_sizes, int n_in,
                              void* d_out, int out_size, void* d_ws, size_t ws_size,
                              hipStream_t stream) {
  const float* pos   = (const float*)d_in[0];
  const int*   bidcs = (const int*)  d_in[1];
  const float* beq   = (const float*)d_in[2];
  const float* btol  = (const float*)d_in[3];
  const int*   aidcs = (const int*)  d_in[4];
  const float* aeq   = (const float*)d_in[5];
  const float* atol  = (const float*)d_in[6];
  const int*   didcs = (const int*)  d_in[7];
  const float* deq   = (const float*)d_in[8];
  const int*   mov   = (const int*)  d_in[9];
  const float* dtau  = (const float*)d_in[10];

  int natoms = in_sizes[0] / 3;
  int nb = in_sizes[2], na = in_sizes[5], nd = in_sizes[8], nm = in_sizes[9];

  float* out  = (float*)d_out;
  float* gacc = out + (size_t)3 * natoms;   // gradient region doubles as accumulator
  float* esum = (float*)d_ws;               // 3*ESLOTS floats of scratch

  float bK = 1000.0f / (float)nb;
  float aK = 150.0f  / (float)na;
  float dK = 1.0f    / (float)nd;

  hipMemsetAsync(gacc, 0, (size_t)3 * natoms * sizeof(float), stream);
  hipMemsetAsync(esum, 0, 3 * ESLOTS * sizeof(float), stream);

  dim3 B(BLOCK);
  k_bonds <<<(nb + BLOCK - 1) / BLOCK, B, 0, stream>>>(pos, bidcs, beq, btol, gacc, esum, nb, bK);
  k_angles<<<(na + BLOCK - 1) / BLOCK, B, 0, stream>>>(pos, aidcs, aeq, atol, gacc, esum, na, aK);
  k_dihs  <<<(nd + BLOCK - 1) / BLOCK, B, 0, stream>>>(pos, didcs, deq,       gacc, esum, nd, dK);
  k_finalize<<<(natoms + BLOCK - 1) / BLOCK, B, 0, stream>>>(pos, esum, dtau, out, natoms, bK, aK, dK);
  k_move  <<<(nm + BLOCK - 1) / BLOCK, B, 0, stream>>>(pos, mov, dtau, out, nm, natoms);
}